// NGCF_71416716198489
// MI455X (gfx1250) — compile-verified
//
#include <hip/hip_runtime.h>
#include <hip/hip_bf16.h>

#define N_USERS 100000
#define N_ITEMS 50000
#define NTOT    (N_USERS + N_ITEMS)   // 150000
#define NNZ_E   6000000
#define D       64
#define OUTD    192                    // 3*D concat layout in d_out
#define LRELU   0.2f

typedef __attribute__((ext_vector_type(2))) float v2f;
typedef __attribute__((ext_vector_type(8))) float v8f;

// ---------------------------------------------------------------------------
// Copy ego embeddings (users then items) into columns [0,64) of d_out [N,192]
// ---------------------------------------------------------------------------
__global__ void k_init_h(const float* __restrict__ ue,
                         const float* __restrict__ ie,
                         float* __restrict__ out) {
  int idx = blockIdx.x * 256 + threadIdx.x;
  if (idx >= NTOT * D) return;
  int row = idx >> 6;
  int d   = idx & 63;
  float v = (row < N_USERS) ? ue[row * D + d] : ie[(row - N_USERS) * D + d];
  out[(size_t)row * OUTD + d] = v;
}

// ---------------------------------------------------------------------------
// Zero the scatter accumulator x [N,64]
// ---------------------------------------------------------------------------
__global__ void k_zero(float* __restrict__ x) {
  int idx = blockIdx.x * 256 + threadIdx.x;
  if (idx < NTOT * D) x[idx] = 0.0f;
}

// ---------------------------------------------------------------------------
// SpMM scatter: x[rows[e]] += vals[e] * h[cols[e]]
// One wave (32 lanes) per edge; each lane handles 2 consecutive floats
// -> 256B coalesced gather per edge row (L2-resident), 2 f32 atomics per lane.
// h has row stride OUTD (it is a column slice of d_out); x has stride D.
// ---------------------------------------------------------------------------
__global__ void k_spmm(const int*   __restrict__ rows,
                       const int*   __restrict__ cols,
                       const float* __restrict__ vals,
                       const float* __restrict__ h,
                       float*       __restrict__ x) {
  int wid  = (blockIdx.x * blockDim.x + threadIdx.x) >> 5;
  int lane = threadIdx.x & 31;
  if (wid >= NNZ_E) return;
  int   r = rows[wid];
  int   c = cols[wid];
  float v = vals[wid];
  const float2 hv = *(const float2*)(h + (size_t)c * OUTD + 2 * lane);
  float* dst = x + (size_t)r * D + 2 * lane;
  atomicAdd(dst + 0, v * hv.x);
  atomicAdd(dst + 1, v * hv.y);
}

// ---------------------------------------------------------------------------
// Fused dense step using V_WMMA_F32_16X16X4_F32 (fp32 matrix pipe):
//   out = leaky_relu( (h + x) @ W1 + (x * h) @ W2 + b1 + b2 )
// One wave per 16-row tile. K=64 stepped by 4; N covered by four 16-wide tiles.
// A fragment (16x4 f32, 2 VGPRs): lane = {m = lane&15, khalf = lane>>4},
//   a[v] = A[m][k0 + 2*khalf + v].
// B fragment (4x16 f32, 2 VGPRs): lane n = lane&15,
//   b[v] = W[(k0 + 2*khalf + v)][ntile*16 + n].
// C/D (16x16 f32, 8 VGPRs): acc[r] = D[r + 8*(lane>>4)][ntile*16 + (lane&15)].
// Bias depends only on the column -> fold into accumulator init.
// ---------------------------------------------------------------------------
__global__ void k_dense(const float* __restrict__ h,    // stride OUTD
                        const float* __restrict__ x,    // stride D
                        const float* __restrict__ W1,   // [D,D] row-major
                        const float* __restrict__ b1,   // [D]
                        const float* __restrict__ W2,   // [D,D] row-major
                        const float* __restrict__ b2,   // [D]
                        float*       __restrict__ outp) // stride OUTD
{
  int wave = threadIdx.x >> 5;              // 4 waves / block
  int lane = threadIdx.x & 31;
  int tile = blockIdx.x * 4 + wave;         // 16-row tile index
  int row0 = tile * 16;
  if (row0 >= NTOT) return;                 // wave-uniform exit (EXEC stays full)

  int m  = lane & 15;
  int kh = lane >> 4;                       // which K-half of the fragment
  int n  = lane & 15;

  v8f acc[4];
  for (int t = 0; t < 4; ++t) {
    float bias = b1[t * 16 + n] + b2[t * 16 + n];
    for (int r = 0; r < 8; ++r) acc[t][r] = bias;
  }

  const float* hrow = h + (size_t)(row0 + m) * OUTD;
  const float* xrow = x + (size_t)(row0 + m) * D;

  for (int k0 = 0; k0 < D; k0 += 4) {
    int ka = k0 + 2 * kh;
    float h0 = hrow[ka], h1 = hrow[ka + 1];
    float x0 = xrow[ka], x1 = xrow[ka + 1];
    v2f a1; a1[0] = h0 + x0; a1[1] = h1 + x1;   // (h + x) fragment
    v2f a2; a2[0] = h0 * x0; a2[1] = h1 * x1;   // (x * h) fragment
#pragma unroll
    for (int t = 0; t < 4; ++t) {
      v2f bw1, bw2;
      bw1[0] = W1[(ka + 0) * D + t * 16 + n];
      bw1[1] = W1[(ka + 1) * D + t * 16 + n];
      bw2[0] = W2[(ka + 0) * D + t * 16 + n];
      bw2[1] = W2[(ka + 1) * D + t * 16 + n];
      acc[t] = __builtin_amdgcn_wmma_f32_16x16x4_f32(
          false, a1, false, bw1, (short)0, acc[t], false, false);
      acc[t] = __builtin_amdgcn_wmma_f32_16x16x4_f32(
          false, a2, false, bw2, (short)0, acc[t], false, false);
    }
  }

  for (int t = 0; t < 4; ++t) {
    for (int r = 0; r < 8; ++r) {
      int M = r + 8 * kh;
      float v = acc[t][r];
      v = (v > 0.0f) ? v : (LRELU * v);
      outp[(size_t)(row0 + M) * OUTD + t * 16 + n] = v;
    }
  }
}

// ---------------------------------------------------------------------------
// Row L2-normalize in place (one wave32 per row, 2 floats per lane)
// ---------------------------------------------------------------------------
__global__ void k_norm(float* __restrict__ hcol) {     // stride OUTD
  int wid  = (blockIdx.x * blockDim.x + threadIdx.x) >> 5;
  int lane = threadIdx.x & 31;
  if (wid >= NTOT) return;
  float* p = hcol + (size_t)wid * OUTD + 2 * lane;
  float2 v = *(float2*)p;
  float s = v.x * v.x + v.y * v.y;
#pragma unroll
  for (int off = 16; off > 0; off >>= 1) s += __shfl_xor(s, off, 32);
  float nrm = sqrtf(s);
  nrm = fmaxf(nrm, 1e-12f);
  float inv = 1.0f / nrm;
  v.x *= inv; v.y *= inv;
  *(float2*)p = v;
}

// ---------------------------------------------------------------------------
// Launch: inputs are rows, cols, vals, user_emb, item_emb, W1, b1, W2, b2.
// d_out is [N,192] = concat(h0,h1,h2); each layer's h lives in its own
// 64-column slice of d_out. Workspace holds only x [N,64] (38.4 MB).
// ---------------------------------------------------------------------------
extern "C" void kernel_launch(void* const* d_in, const int* in_sizes, int n_in,
                              void* d_out, int out_size, void* d_ws, size_t ws_size,
                              hipStream_t stream) {
  const int*   rows = (const int*)d_in[0];
  const int*   cols = (const int*)d_in[1];
  const float* vals = (const float*)d_in[2];
  const float* ue   = (const float*)d_in[3];
  const float* ie   = (const float*)d_in[4];
  const float* W1   = (const float*)d_in[5];  // [2,64,64]
  const float* b1   = (const float*)d_in[6];  // [2,64]
  const float* W2   = (const float*)d_in[7];
  const float* b2   = (const float*)d_in[8];
  float* out = (float*)d_out;
  float* x   = (float*)d_ws;                  // [N,64] scatter accumulator

  (void)in_sizes; (void)n_in; (void)out_size; (void)ws_size;

  k_init_h<<<(NTOT * D + 255) / 256, 256, 0, stream>>>(ue, ie, out);

  for (int l = 0; l < 2; ++l) {
    k_zero<<<(NTOT * D + 255) / 256, 256, 0, stream>>>(x);
    // 8 edges per 256-thread block (one wave per edge)
    k_spmm<<<(NNZ_E + 7) / 8, 256, 0, stream>>>(rows, cols, vals,
                                                out + l * D, x);
    // 9375 16-row tiles, 4 waves (tiles) per block
    k_dense<<<(NTOT / 16 + 3) / 4, 128, 0, stream>>>(
        out + l * D, x,
        W1 + (size_t)l * D * D, b1 + (size_t)l * D,
        W2 + (size_t)l * D * D, b2 + (size_t)l * D,
        out + (l + 1) * D);
    // one wave per row
    k_norm<<<(NTOT * 32 + 255) / 256, 256, 0, stream>>>(out + (l + 1) * D);
  }
}